// ModelNew_69879117906290
// MI455X (gfx1250) — compile-verified
//
#include <hip/hip_runtime.h>

typedef float v2f __attribute__((ext_vector_type(2)));
typedef float v8f __attribute__((ext_vector_type(8)));

#define BSZ     16
#define SEQ     8192
#define NH      8
#define PDIM    64
#define NDIM    16
#define LCHUNK  64
#define NCHUNK  128            // SEQ / LCHUNK
#define GROUPS  16
#define CPG     (NCHUNK / GROUPS)
#define LSTRIDE 66             // padded LDS stride (even -> 8B-aligned pairs, conflict-free)

// ---------------------------------------------------------------------------
// Pass 1: per-chunk log-decay sums + inclusive suffix scan over chunks.
// logsuf[bh*NCHUNK + c] = sum_{c'=c..C-1} sum_l A[b, c'*L + l, h]
// ---------------------------------------------------------------------------
__global__ void k_chunk_scan(const float* __restrict__ A, float* __restrict__ logsuf) {
    int bh = blockIdx.x;               // 0..BSZ*NH-1
    int b  = bh / NH, h = bh % NH;
    int c  = threadIdx.x;              // 0..NCHUNK-1

    const float* Ap = A + ((size_t)(b * SEQ + c * LCHUNK)) * NH + h;
    float s = 0.f;
    #pragma unroll 8
    for (int l = 0; l < LCHUNK; ++l) s += Ap[(size_t)l * NH];

    __shared__ float buf[NCHUNK];
    buf[c] = s;
    __syncthreads();
    // Hillis-Steele inclusive suffix scan
    for (int off = 1; off < NCHUNK; off <<= 1) {
        float v = (c + off < NCHUNK) ? buf[c + off] : 0.f;
        __syncthreads();
        buf[c] += v;
        __syncthreads();
    }
    logsuf[(size_t)bh * NCHUNK + c] = buf[c];
}

// ---------------------------------------------------------------------------
// Pass 2: out = initial_states * exp(total log decay)
// ---------------------------------------------------------------------------
__global__ void k_init_out(const float* __restrict__ init,
                           const float* __restrict__ logsuf,
                           float* __restrict__ out) {
    int i = blockIdx.x * blockDim.x + threadIdx.x;
    if (i >= BSZ * NH * NDIM * PDIM) return;
    int bh = i / (NDIM * PDIM);
    out[i] = init[i] * __expf(logsuf[(size_t)bh * NCHUNK]);
}

// ---------------------------------------------------------------------------
// Pass 3: weighted chunk GEMMs via V_WMMA_F32_16X16X4_F32, atomically
// accumulated into out. One block = one (b,h) pair x one group of 8 chunks.
// 4 waves per block, each owning a 16-wide P tile of the 16x64 output.
// ---------------------------------------------------------------------------
__global__ void __launch_bounds__(128) k_chunk_gemm(
        const float* __restrict__ X, const float* __restrict__ A,
        const float* __restrict__ Bm, const float* __restrict__ logsuf,
        float* __restrict__ out) {
    __shared__ float sA[LCHUNK];             // cumsum(A) within chunk
    __shared__ float sW[LCHUNK];             // per-position weights
    __shared__ float sX[PDIM * LSTRIDE];     // X transposed: sX[p*66 + l]
    __shared__ float sB[NDIM * LSTRIDE];     // w*B transposed: sB[n*66 + l]

    int blk = blockIdx.x;
    int g   = blk % GROUPS;
    int bh  = blk / GROUPS;
    int b   = bh / NH, h = bh % NH;

    int t    = threadIdx.x;
    int lane = t & 31;
    int wave = t >> 5;
    int col  = lane & 15;
    int hi   = (lane >> 4) & 1;              // 0: lanes 0-15 (K=0,1), 1: lanes 16-31 (K=2,3)
    int p0   = wave * 16;

    v8f acc = {};

    for (int c = g * CPG; c < (g + 1) * CPG; ++c) {
        __syncthreads();                     // protect LDS reuse across iterations

        // --- load A chunk ---
        if (t < LCHUNK)
            sA[t] = A[((size_t)(b * SEQ + c * LCHUNK + t)) * NH + h];
        __syncthreads();

        // --- inclusive prefix scan (A_cs) over 64 positions ---
        #pragma unroll
        for (int off = 1; off < LCHUNK; off <<= 1) {
            float v = 0.f;
            if (t < LCHUNK && t >= off) v = sA[t - off];
            __syncthreads();
            if (t < LCHUNK) sA[t] += v;
            __syncthreads();
        }

        // --- combined weight: exp(logsuffix[c] - A_cs[l]) ---
        float ls = logsuf[(size_t)bh * NCHUNK + c];
        if (t < LCHUNK) sW[t] = __expf(ls - sA[t]);

        // --- X chunk (64x64) -> LDS, transposed, coalesced float4 loads ---
        const float* Xp = X + ((size_t)(b * SEQ + c * LCHUNK)) * NH * PDIM + (size_t)h * PDIM;
        #pragma unroll
        for (int i = 0; i < 8; ++i) {
            int idx4 = t + i * 128;          // 1024 float4s
            int l    = idx4 >> 4;
            int p4   = (idx4 & 15) << 2;
            float4 v = *(const float4*)(Xp + (size_t)l * NH * PDIM + p4);
            sX[(p4 + 0) * LSTRIDE + l] = v.x;
            sX[(p4 + 1) * LSTRIDE + l] = v.y;
            sX[(p4 + 2) * LSTRIDE + l] = v.z;
            sX[(p4 + 3) * LSTRIDE + l] = v.w;
        }
        __syncthreads();                     // sW ready before scaling B

        // --- B chunk (64x16), scaled by w[l] -> LDS transposed ---
        const float* Bp = Bm + ((size_t)(b * SEQ + c * LCHUNK)) * NH * NDIM + (size_t)h * NDIM;
        #pragma unroll
        for (int i = 0; i < 8; ++i) {
            int e = t + i * 128;             // 1024 elements
            int l = e >> 4;
            int n = e & 15;
            sB[n * LSTRIDE + l] = sW[l] * Bp[(size_t)l * NH * NDIM + n];
        }
        __syncthreads();

        // --- 16 x V_WMMA_F32_16X16X4_F32: D(16n x 16p) += Bd^T * X ---
        // A frag (16x4, M=n): lane<16 -> K=0,1 ; lane>=16 -> K=2,3 ; M = col
        // B frag (4x16, N=p): lane<16 -> K=0,1 ; lane>=16 -> K=2,3 ; N = p0+col
        #pragma unroll
        for (int k = 0; k < 16; ++k) {
            int koff = 4 * k + 2 * hi;
            v2f af, bf;
            af.x = sB[col * LSTRIDE + koff];
            af.y = sB[col * LSTRIDE + koff + 1];
            bf.x = sX[(p0 + col) * LSTRIDE + koff];
            bf.y = sX[(p0 + col) * LSTRIDE + koff + 1];
            acc = __builtin_amdgcn_wmma_f32_16x16x4_f32(
                /*neg_a=*/false, af, /*neg_b=*/false, bf,
                /*c_mod=*/(short)0, acc, /*reuse_a=*/false, /*reuse_b=*/false);
        }
    }

    // --- C/D layout: VGPR r -> M = r (lanes 0-15) / r+8 (lanes 16-31), N = col ---
    float* outp = out + (size_t)bh * NDIM * PDIM;
    #pragma unroll
    for (int r = 0; r < 8; ++r) {
        int n = r + 8 * hi;
        int p = p0 + col;
        atomicAdd(&outp[n * PDIM + p], acc[r]);
    }
}

// ---------------------------------------------------------------------------
extern "C" void kernel_launch(void* const* d_in, const int* in_sizes, int n_in,
                              void* d_out, int out_size, void* d_ws, size_t ws_size,
                              hipStream_t stream) {
    const float* X    = (const float*)d_in[0];   // (16,8192,8,64)
    const float* A    = (const float*)d_in[1];   // (16,8192,8)
    const float* Bm   = (const float*)d_in[2];   // (16,8192,8,16)
    const float* init = (const float*)d_in[3];   // (16,8,16,64)
    float* out    = (float*)d_out;               // (16,8,16,64)
    float* logsuf = (float*)d_ws;                // BSZ*NH*NCHUNK floats = 64KB

    k_chunk_scan<<<BSZ * NH, NCHUNK, 0, stream>>>(A, logsuf);
    int tot = BSZ * NH * NDIM * PDIM;
    k_init_out<<<(tot + 255) / 256, 256, 0, stream>>>(init, logsuf, out);
    k_chunk_gemm<<<BSZ * NH * GROUPS, 128, 0, stream>>>(X, A, Bm, logsuf, out);
}